// GCNLayer_7327214207511
// MI455X (gfx1250) — compile-verified
//
#include <hip/hip_runtime.h>

#define D 128
#define MROWS 32
#define PAD 260   // 256 cols + 4 pad -> row stride rotates 4 banks, conflict-free ds_load_b64

typedef float v2f __attribute__((ext_vector_type(2)));
typedef float v8f __attribute__((ext_vector_type(8)));
typedef int   v4i __attribute__((ext_vector_type(4)));

#if defined(__has_builtin)
#if __has_builtin(__builtin_amdgcn_global_load_async_to_lds_b128)
#define HAVE_ASYNC_LDS 1
#endif
#endif

__global__ void zero_ws_kernel(float* __restrict__ p, int n) {
    int i = blockIdx.x * blockDim.x + threadIdx.x;
    if (i < n) p[i] = 0.0f;
}

__global__ void degree_kernel(const int* __restrict__ src, float* __restrict__ deg, int E) {
    int e = blockIdx.x * blockDim.x + threadIdx.x;
    if (e < E) unsafeAtomicAdd(&deg[src[e]], 1.0f);   // native global_atomic_add_f32
}

__global__ void dis_kernel(const float* __restrict__ deg, float* __restrict__ dis, int N) {
    int i = blockIdx.x * blockDim.x + threadIdx.x;
    if (i < N) {
        float d = deg[i];
        dis[i] = (d > 0.0f) ? rsqrtf(d) : 0.0f;
    }
}

// One wave32 per edge: each lane moves one float4 (32 lanes x 4 = 128 floats),
// scaled gather from L2-resident feats, native f32 atomic scatter into rst.
__global__ void scatter_kernel(const float* __restrict__ feats,
                               const int* __restrict__ src,
                               const int* __restrict__ dst,
                               const float* __restrict__ dis,
                               float* __restrict__ rst, int E) {
    int wave = (blockIdx.x * blockDim.x + threadIdx.x) >> 5;
    int lane = threadIdx.x & 31;
    if (wave >= E) return;
    int s = src[wave];
    int d = dst[wave];
    float sc = dis[s];
    float4 v = ((const float4*)(feats + (size_t)s * D))[lane];
    float* outp = rst + (size_t)d * D + lane * 4;
    unsafeAtomicAdd(outp + 0, v.x * sc);
    unsafeAtomicAdd(outp + 1, v.y * sc);
    unsafeAtomicAdd(outp + 2, v.z * sc);
    unsafeAtomicAdd(outp + 3, v.w * sc);
}

// Fused dual GEMM: out = (rst*dis) @ Wmsg^T + feats @ Wskip^T + (bmsg + bskip)
// A tile = [32 rows][256 cols]  (cols 0..127 = rst*dis, 128..255 = feats) in LDS.
// feats half staged with GLOBAL_LOAD_ASYNC_TO_LDS_B128 (ASYNCcnt), overlapped
// with the VALU-scaled rst half. 8 waves/block; wave w owns output cols
// [w*16, w*16+16), two 16-row WMMA tiles.
__global__ void __launch_bounds__(256)
gemm_kernel(const float* __restrict__ feats,
            const float* __restrict__ rst,
            const float* __restrict__ dis,
            const float* __restrict__ Wmsg,
            const float* __restrict__ bmsg,
            const float* __restrict__ Wskip,
            const float* __restrict__ bskip,
            float* __restrict__ out, int N) {
    __shared__ float tileA[MROWS * PAD];

    const int t = threadIdx.x;
    const int rowbase = blockIdx.x * MROWS;

#ifdef HAVE_ASYNC_LDS
    // Async stage of the skip half: 32 rows x 128 floats = 1024 b128 chunks.
    for (int it = 0; it < 4; ++it) {
        int idx = t + it * 256;      // float4 chunk id
        int row = idx >> 5;          // 0..31
        int c4  = idx & 31;          // 0..31 float4s within a row
        const float* g = feats + (size_t)(rowbase + row) * D + c4 * 4;
        __builtin_amdgcn_global_load_async_to_lds_b128(
            (__attribute__((address_space(1))) v4i*)g,
            (__attribute__((address_space(3))) v4i*)&tileA[row * PAD + D + c4 * 4],
            0, 0);
    }
#endif

    // VALU-scaled msg half: 32 rows x 128 floats, 16 per thread.
    for (int i = 0; i < 16; ++i) {
        int idx = t + i * 256;
        int row = idx >> 7;          // 0..31
        int c   = idx & 127;
        tileA[row * PAD + c] = rst[(size_t)(rowbase + row) * D + c] * dis[rowbase + row];
    }

#ifdef HAVE_ASYNC_LDS
#if defined(__has_builtin) && __has_builtin(__builtin_amdgcn_s_wait_asynccnt)
    __builtin_amdgcn_s_wait_asynccnt(0);
#else
    asm volatile("s_wait_asynccnt 0" ::: "memory");
#endif
#else
    for (int i = 0; i < 16; ++i) {
        int idx = t + i * 256;
        int row = idx >> 7;
        int c   = idx & 127;
        tileA[row * PAD + D + c] = feats[(size_t)(rowbase + row) * D + c];
    }
#endif
    __syncthreads();

    const int wave = t >> 5;
    const int lane = t & 31;
    const int n  = lane & 15;   // col within 16-wide tile / A row (lanes 0-15 & 16-31 both M=0..15)
    const int kq = lane >> 4;   // selects K pair {0,1} vs {2,3} of the 16x16x4 fragment
    const int col = wave * 16 + n;

    v8f c0 = {};  // rows rowbase+0..15
    v8f c1 = {};  // rows rowbase+16..31

    for (int srcm = 0; srcm < 2; ++srcm) {
        const float* __restrict__ W = srcm ? Wskip : Wmsg;
        const int base = srcm * D;
        for (int k = 0; k < D; k += 4) {
            // B fragment: B[k][n] = W[col][k]  (W is [out][in], row-major)
            v2f b = *(const v2f*)(W + (size_t)col * D + k + 2 * kq);
            // A fragments from LDS (8B-aligned, PAD even)
            v2f a0 = *(const v2f*)(&tileA[n * PAD + base + k + 2 * kq]);
            v2f a1 = *(const v2f*)(&tileA[(n + 16) * PAD + base + k + 2 * kq]);
            c0 = __builtin_amdgcn_wmma_f32_16x16x4_f32(false, a0, false, b,
                                                       (short)0, c0, false, false);
            c1 = __builtin_amdgcn_wmma_f32_16x16x4_f32(false, a1, false, b,
                                                       (short)0, c1, false, false);
        }
    }

    const float bias = bmsg[col] + bskip[col];
    // C/D layout: VGPR j -> M = j + kq*8 ; N = lane&15
    for (int j = 0; j < 8; ++j) {
        out[(size_t)(rowbase + kq * 8 + j) * D + col]      = c0[j] + bias;
        out[(size_t)(rowbase + 16 + kq * 8 + j) * D + col] = c1[j] + bias;
    }
}

extern "C" void kernel_launch(void* const* d_in, const int* in_sizes, int n_in,
                              void* d_out, int out_size, void* d_ws, size_t ws_size,
                              hipStream_t stream) {
    const float* feats = (const float*)d_in[0];
    const int*   src   = (const int*)d_in[1];
    const int*   dst   = (const int*)d_in[2];
    const float* Wskip = (const float*)d_in[3];
    const float* bskip = (const float*)d_in[4];
    const float* Wmsg  = (const float*)d_in[5];
    const float* bmsg  = (const float*)d_in[6];
    float* out = (float*)d_out;

    const int N = in_sizes[0] / D;   // 100000
    const int E = in_sizes[1];       // 640000

    // Workspace layout: [deg: N][dis: N][rst: N*D]  (~52 MB)
    float* deg = (float*)d_ws;
    float* dis = deg + N;
    float* rst = dis + N;

    const int zn = N * (D + 2);
    zero_ws_kernel<<<(zn + 255) / 256, 256, 0, stream>>>(deg, zn);
    degree_kernel<<<(E + 255) / 256, 256, 0, stream>>>(src, deg, E);
    dis_kernel<<<(N + 255) / 256, 256, 0, stream>>>(deg, dis, N);
    scatter_kernel<<<(E + 7) / 8, 256, 0, stream>>>(feats, src, dst, dis, rst, E);
    gemm_kernel<<<(N + MROWS - 1) / MROWS, 256, 0, stream>>>(
        feats, rst, dis, Wmsg, bmsg, Wskip, bskip, out, N);
}